// SOCA_71107478552742
// MI455X (gfx1250) — compile-verified
//
#include <hip/hip_runtime.h>
#include <math.h>

typedef float v2f __attribute__((ext_vector_type(2)));
typedef float v8f __attribute__((ext_vector_type(8)));
typedef int   vi4 __attribute__((vector_size(16)));   // matches builtin param type

#define B_  32
#define C_  256
#define M_  4096
#define LDT 36   // padded LDS row stride (dwords): even (8B-aligned v2f) and /4 (16B float4)

#if __has_builtin(__builtin_amdgcn_global_load_async_to_lds_b128) && \
    __has_builtin(__builtin_amdgcn_global_load_async_to_lds_b32)  && \
    __has_builtin(__builtin_amdgcn_s_wait_asynccnt)
#define SOCA_ASYNC 1
#else
#define SOCA_ASYNC 0
#endif

#define AS1 __attribute__((address_space(1)))
#define AS3 __attribute__((address_space(3)))

__device__ __forceinline__ v8f wmma4(v2f a, v2f b, v8f c) {
  // D = A(16x4,f32) * B(4x16,f32) + C(16x16,f32)
  return __builtin_amdgcn_wmma_f32_16x16x4_f32(false, a, false, b, (short)0, c, false, false);
}

__device__ __forceinline__ void tile_ld128(const float* g, float* l) {
#if SOCA_ASYNC
  __builtin_amdgcn_global_load_async_to_lds_b128((AS1 vi4*)g, (AS3 vi4*)l, 0, 0);
#else
  *(float4*)l = *(const float4*)g;
#endif
}

__device__ __forceinline__ void tile_ld32(const float* g, float* l) {
#if SOCA_ASYNC
  __builtin_amdgcn_global_load_async_to_lds_b32((AS1 int*)g, (AS3 int*)l, 0, 0);
#else
  *l = *g;
#endif
}

__device__ __forceinline__ void tile_wait() {
#if SOCA_ASYNC
  __builtin_amdgcn_s_wait_asynccnt(0);
#endif
}

// ---------------- per-channel mean: mu[b*C+c] = (1/4096) * sum_m x ----------------
__global__ __launch_bounds__(256) void soca_mean(const float* __restrict__ x,
                                                 float* __restrict__ mu) {
  int row = blockIdx.x;                         // b*C + c, 8192 rows
  const float4* xr = (const float4*)(x + ((size_t)row << 12));
  int t = threadIdx.x;
  float acc = 0.f;
#pragma unroll
  for (int i = 0; i < 4; ++i) {
    float4 v = xr[t + (i << 8)];
    acc += (v.x + v.y) + (v.z + v.w);
  }
  __shared__ float red[256];
  red[t] = acc; __syncthreads();
  for (int s = 128; s > 0; s >>= 1) { if (t < s) red[t] += red[t + s]; __syncthreads(); }
  if (t == 0) mu[row] = red[0] * (1.f / 4096.f);
}

// ---------------- covariance via WMMA: cov = X.Xt/M - mu.mu^T ----------------
__global__ __launch_bounds__(256) void soca_gram(const float* __restrict__ X,
                                                 const float* __restrict__ mu,
                                                 float* __restrict__ Cov) {
  __shared__ float lA[64 * LDT];
  __shared__ float lB[64 * LDT];
  int b   = blockIdx.x >> 4;
  int til = blockIdx.x & 15;
  int mb = (til >> 2) << 6;
  int nb = (til & 3) << 6;
  const float* Xb = X + (size_t)b * C_ * M_;
  int t = threadIdx.x, lane = t & 31, wave = t >> 5;
  int lmb = (wave & 3) << 4;      // wave's 16-row block in tile
  int lnb = (wave >> 2) << 5;     // wave's 32-col block in tile
  v8f acc0 = {}; v8f acc1 = {};
  for (int kk = 0; kk < M_; kk += 32) {
#pragma unroll
    for (int i = 0; i < 2; ++i) {               // 64x32 tile, 16B async Global->LDS
      int f4 = t + (i << 8);
      int r  = f4 >> 3;
      int c4 = (f4 & 7) << 2;
      tile_ld128(Xb + (size_t)(mb + r) * M_ + kk + c4, &lA[r * LDT + c4]);
      tile_ld128(Xb + (size_t)(nb + r) * M_ + kk + c4, &lB[r * LDT + c4]);
    }
    tile_wait();
    __syncthreads();
    int mr = lmb + (lane & 15);
    int nr = lnb + (lane & 15);
    int kh = (lane >> 4) << 1;                  // lanes 16..31 hold K2,K3
#pragma unroll
    for (int ks = 0; ks < 32; ks += 4) {
      int k0 = ks + kh;
      v2f a  = *(const v2f*)&lA[mr * LDT + k0];
      v2f b0 = *(const v2f*)&lB[nr * LDT + k0];
      v2f b1 = *(const v2f*)&lB[(nr + 16) * LDT + k0];
      acc0 = wmma4(a, b0, acc0);
      acc1 = wmma4(a, b1, acc1);
    }
    __syncthreads();
  }
  const float* mub = mu + b * C_;
  float* Cb = Cov + ((size_t)b << 16);
  int m0 = mb + lmb + ((lane >> 4) << 3);       // C/D layout: vgpr r -> M = r (+8 for hi lanes)
  int n0 = nb + lnb + (lane & 15);
  int n1 = n0 + 16;
  float mn0 = mub[n0], mn1 = mub[n1];
#pragma unroll
  for (int r = 0; r < 8; ++r) {
    int m = m0 + r;
    float mm = mub[m];
    Cb[((size_t)m << 8) + n0] = acc0[r] * (1.f / (float)M_) - mm * mn0;
    Cb[((size_t)m << 8) + n1] = acc1[r] * (1.f / (float)M_) - mm * mn1;
  }
}

// ---------------- trace / normalizers ----------------
__global__ __launch_bounds__(256) void soca_trace(const float* __restrict__ Cov,
                                                  float* __restrict__ norm,
                                                  float* __restrict__ rnorm,
                                                  float* __restrict__ snorm) {
  __shared__ float red[256];
  int b = blockIdx.x, t = threadIdx.x;
  red[t] = Cov[((size_t)b << 16) + (size_t)t * 257];
  __syncthreads();
  for (int s = 128; s > 0; s >>= 1) { if (t < s) red[t] += red[t + s]; __syncthreads(); }
  if (t == 0) { float v = red[0]; norm[b] = v; rnorm[b] = 1.f / v; snorm[b] = sqrtf(v); }
}

// ---------------- A = cov/normA (in place); Z0 = 0.5*(3I - A) ----------------
__global__ __launch_bounds__(256) void soca_prep(float* __restrict__ A,
                                                 const float* __restrict__ rnorm,
                                                 float* __restrict__ Z) {
  int i = blockIdx.x * 256 + threadIdx.x;       // < 32*256*256 = 2^21
  int b = i >> 16;
  int rc = i & 65535;
  int r = rc >> 8, c = rc & 255;
  float a = A[i] * rnorm[b];
  A[i] = a;
  Z[i] = ((r == c) ? 1.5f : 0.f) - 0.5f * a;
}

// ---------------- batched 256x256x256 WMMA GEMM with NS epilogues ----------------
// mode 0: D = A@B ; mode 1: D = 0.5*(3I - A@B) ; mode 2: D = (A@B)*scale[b]
__global__ __launch_bounds__(256) void soca_ns_gemm(const float* __restrict__ Am,
                                                    const float* __restrict__ Bm,
                                                    float* __restrict__ Dm,
                                                    const float* __restrict__ scale,
                                                    int mode) {
  __shared__ float lA[64 * LDT];
  __shared__ float lB[64 * LDT];   // stored transposed: lB[n][k]
  int b   = blockIdx.x >> 4;
  int til = blockIdx.x & 15;
  int mb = (til >> 2) << 6;
  int nb = (til & 3) << 6;
  const float* Ab = Am + ((size_t)b << 16);
  const float* Bb = Bm + ((size_t)b << 16);
  float*       Db = Dm + ((size_t)b << 16);
  int t = threadIdx.x, lane = t & 31, wave = t >> 5;
  int lmb = (wave & 3) << 4;
  int lnb = (wave >> 2) << 5;
  v8f acc0 = {}; v8f acc1 = {};
  for (int kk = 0; kk < C_; kk += 32) {
#pragma unroll
    for (int i = 0; i < 2; ++i) {               // A tile 64x32, direct (async B128)
      int f4 = t + (i << 8);
      int r  = f4 >> 3;
      int c4 = (f4 & 7) << 2;
      tile_ld128(Ab + ((size_t)(mb + r) << 8) + kk + c4, &lA[r * LDT + c4]);
    }
#pragma unroll
    for (int i = 0; i < 8; ++i) {               // B tile 32x64 -> transpose to [n][k] (async B32)
      int e = t + (i << 8);
      int k = e >> 6;
      int n = e & 63;
      tile_ld32(Bb + ((size_t)(kk + k) << 8) + nb + n, &lB[n * LDT + k]);
    }
    tile_wait();
    __syncthreads();
    int mr = lmb + (lane & 15);
    int nr = lnb + (lane & 15);
    int kh = (lane >> 4) << 1;
#pragma unroll
    for (int ks = 0; ks < 32; ks += 4) {
      int k0 = ks + kh;
      v2f a  = *(const v2f*)&lA[mr * LDT + k0];
      v2f b0 = *(const v2f*)&lB[nr * LDT + k0];
      v2f b1 = *(const v2f*)&lB[(nr + 16) * LDT + k0];
      acc0 = wmma4(a, b0, acc0);
      acc1 = wmma4(a, b1, acc1);
    }
    __syncthreads();
  }
  int m0 = mb + lmb + ((lane >> 4) << 3);
  int n0 = nb + lnb + (lane & 15);
  int n1 = n0 + 16;
  float sc = (mode == 2) ? scale[b] : 1.f;
#pragma unroll
  for (int r = 0; r < 8; ++r) {
    int m = m0 + r;
    float d0 = acc0[r], d1 = acc1[r];
    if (mode == 1) {
      d0 = ((m == n0) ? 1.5f : 0.f) - 0.5f * d0;
      d1 = ((m == n1) ? 1.5f : 0.f) - 0.5f * d1;
    } else {
      d0 *= sc; d1 *= sc;
    }
    Db[((size_t)m << 8) + n0] = d0;
    Db[((size_t)m << 8) + n1] = d1;
  }
}

// ---------------- s[b,c] = mean over rows i of S[b,i,c] ----------------
__global__ __launch_bounds__(256) void soca_colmean(const float* __restrict__ S,
                                                    float* __restrict__ s) {
  int b = blockIdx.x, c = threadIdx.x;
  const float* Sb = S + ((size_t)b << 16);
  float acc = 0.f;
  for (int i = 0; i < 256; ++i) acc += Sb[(i << 8) + c];
  s[b * 256 + c] = acc * (1.f / 256.f);
}

// ---------------- gate = sigmoid(w2 @ relu(w1 @ s + b1) + b2) ----------------
__global__ __launch_bounds__(256) void soca_gate(const float* __restrict__ s,
                                                 const float* __restrict__ w1,
                                                 const float* __restrict__ b1,
                                                 const float* __restrict__ w2,
                                                 const float* __restrict__ b2,
                                                 float* __restrict__ gate) {
  __shared__ float sv[256];
  __shared__ float hv[32];
  int b = blockIdx.x, t = threadIdx.x;
  sv[t] = s[b * 256 + t];
  __syncthreads();
  if (t < 32) {
    float acc = b1[t];
    for (int c = 0; c < 256; ++c) acc += w1[t * 256 + c] * sv[c];
    hv[t] = fmaxf(acc, 0.f);
  }
  __syncthreads();
  float acc = b2[t];
#pragma unroll
  for (int o = 0; o < 32; ++o) acc += w2[t * 32 + o] * hv[o];
  gate[b * 256 + t] = 1.f / (1.f + expf(-acc));
}

// ---------------- out = gate[b,c] * x ----------------
__global__ __launch_bounds__(256) void soca_scale(const float* __restrict__ x,
                                                  const float* __restrict__ gate,
                                                  float* __restrict__ out) {
  size_t i = (size_t)blockIdx.x * 256 + threadIdx.x;   // float4 index, 8388608 total
  int bc = (int)(i >> 10);                              // 1024 float4 per (b,c)
  float g = gate[bc];
  float4 v = ((const float4*)x)[i];
  float4 o; o.x = g * v.x; o.y = g * v.y; o.z = g * v.z; o.w = g * v.w;
  ((float4*)out)[i] = o;
}

extern "C" void kernel_launch(void* const* d_in, const int* in_sizes, int n_in,
                              void* d_out, int out_size, void* d_ws, size_t ws_size,
                              hipStream_t stream) {
  (void)in_sizes; (void)n_in; (void)out_size; (void)ws_size;
  const float* x  = (const float*)d_in[0];
  const float* w1 = (const float*)d_in[1];
  const float* b1 = (const float*)d_in[2];
  const float* w2 = (const float*)d_in[3];
  const float* b2 = (const float*)d_in[4];
  float* out = (float*)d_out;
  float* W   = (float*)d_ws;

  float* mu    = W;                 // 8192
  float* norm  = W + 8192;          // 32
  float* rnorm = W + 8224;          // 32
  float* snorm = W + 8256;          // 32
  float* smean = W + 8288;          // 8192
  float* gate  = W + 16480;         // 8192
  const size_t MSZ = (size_t)1 << 21;   // one 32x256x256 f32 matrix stack
  float* M0p = W + 32768;
  float* M1p = M0p + MSZ;
  float* M2p = M1p + MSZ;
  float* M3p = M2p + MSZ;
  float* M4p = M3p + MSZ;
  float* M5p = M4p + MSZ;           // total ws ~50.5 MB

  soca_mean <<<8192, 256, 0, stream>>>(x, mu);
  soca_gram <<<512,  256, 0, stream>>>(x, mu, M0p);
  soca_trace<<<32,   256, 0, stream>>>(M0p, norm, rnorm, snorm);
  soca_prep <<<8192, 256, 0, stream>>>(M0p, rnorm, M1p);

  float* A = M0p; float* Z = M1p; float* Y = M2p; float* T = M3p;
  float* Y2 = M4p; float* Z2 = M5p;

  soca_ns_gemm<<<512, 256, 0, stream>>>(A, Z, Y, nullptr, 0);        // Y0 = A@Z0
  for (int it = 0; it < 3; ++it) {                                   // iterN-2 inner steps
    soca_ns_gemm<<<512, 256, 0, stream>>>(Z, Y, T,  nullptr, 1);     // T = 0.5(3I - Z@Y)
    soca_ns_gemm<<<512, 256, 0, stream>>>(Y, T, Y2, nullptr, 0);     // Y' = Y@T
    soca_ns_gemm<<<512, 256, 0, stream>>>(T, Z, Z2, nullptr, 0);     // Z' = T@Z
    float* tmp = Y; Y = Y2; Y2 = tmp;
    tmp = Z; Z = Z2; Z2 = tmp;
  }
  soca_ns_gemm<<<512, 256, 0, stream>>>(Z, Y, T,  nullptr, 1);       // T = 0.5(3I - Z@Y)
  soca_ns_gemm<<<512, 256, 0, stream>>>(Y, T, Y2, snorm,   2);       // cov_sqrt = (Y@T)*sqrt(normA)

  soca_colmean<<<32,    256, 0, stream>>>(Y2, smean);
  soca_gate   <<<32,    256, 0, stream>>>(smean, w1, b1, w2, b2, gate);
  soca_scale  <<<32768, 256, 0, stream>>>(x, gate, out);
}